// Base_37271726195550
// MI455X (gfx1250) — compile-verified
//
#include <hip/hip_runtime.h>

// Problem constants (from reference)
#define BB 32
#define HH 3
#define MM 2048
#define DD 512
#define LL 32
#define TT 2049

#define MTILE_LOGIT 128   // m's per logit block: 8 waves * 16 rows
#define MTILE_OUT   256   // m's per output block

typedef __attribute__((ext_vector_type(2))) float v2f;
typedef __attribute__((ext_vector_type(8))) float v8f;

// ---------------------------------------------------------------------------
// CDNA5 async global->LDS copy (GLOBAL_LOAD_ASYNC_TO_LDS_B128, ASYNCcnt).
// Builtin signature (leaked by clang diagnostic):
//   void __builtin_amdgcn_global_load_async_to_lds_b128(
//       v4i __attribute__((address_space(1)))* src,   // global
//       v4i __attribute__((address_space(3)))* dst,   // LDS
//       int imm_offset, int imm_cpol);
// ---------------------------------------------------------------------------
#if __has_builtin(__builtin_amdgcn_global_load_async_to_lds_b128)
#define HAS_ASYNC_LDS 1
#else
#define HAS_ASYNC_LDS 0
#endif

typedef int __attribute__((vector_size(16))) vi4;
typedef vi4 __attribute__((address_space(1))) * g_vi4_p;
typedef vi4 __attribute__((address_space(3))) * l_vi4_p;

__device__ __forceinline__ void async_copy_b128(void* lds_dst, const void* gsrc)
{
#if HAS_ASYNC_LDS
    __builtin_amdgcn_global_load_async_to_lds_b128(
        (g_vi4_p)gsrc, (l_vi4_p)lds_dst, /*offset=*/0, /*cpol=*/0);
#else
    *(float4*)lds_dst = *(const float4*)gsrc;
#endif
}

__device__ __forceinline__ void async_wait_all()
{
#if HAS_ASYNC_LDS
#if __has_builtin(__builtin_amdgcn_s_wait_asynccnt)
    __builtin_amdgcn_s_wait_asynccnt(0);
#else
    asm volatile("s_wait_asynccnt 0x0" ::: "memory");
#endif
#endif
}

// ---------------------------------------------------------------------------
// u[b,d] = sum_j B_emb[query[b,j], d] * pos_enc[j, d]
// ---------------------------------------------------------------------------
__global__ __launch_bounds__(DD) void embed_kernel(
        const int* __restrict__ query,
        const float* __restrict__ B_emb,
        const float* __restrict__ pos_enc,
        float* __restrict__ u)
{
    const int b = blockIdx.x;
    const int d = threadIdx.x;            // 0..511, coalesced
    __shared__ int toks[LL];
    if (threadIdx.x < LL) toks[threadIdx.x] = query[b * LL + threadIdx.x];
    __syncthreads();
    float acc = 0.f;
#pragma unroll 4
    for (int j = 0; j < LL; ++j)
        acc += B_emb[(size_t)toks[j] * DD + d] * pos_enc[j * DD + d];
    u[b * DD + d] = acc;
}

// ---------------------------------------------------------------------------
// logit[b,m] = dot(key_mems[b,hop,m,:] + TA[hop, rel_time[b,m], :], u[b,:])
// Batched GEMV mapped onto V_WMMA_F32_16X16X4_F32:
//   A (16x4 f32)  = 16 key rows, 4 k's   (lanes 0-15: K=0,1 ; lanes 16-31: K=2,3)
//   B (4x16 f32)  = u[k] broadcast to all 16 columns
//   C (16x16 f32) accumulates; every column equals the dot product.
// ---------------------------------------------------------------------------
__global__ __launch_bounds__(256) void logit_kernel(
        int hop,
        const int* __restrict__ rel_time,
        const float* __restrict__ key_mems,
        const float* __restrict__ TA,
        const float* __restrict__ u,
        float* __restrict__ logits)
{
    const int b    = blockIdx.y;
    const int tid  = threadIdx.x;
    const int lane = tid & 31;
    const int wave = tid >> 5;

    // Stage u[b,:] (2 KB) into LDS via the async path.
    __shared__ float s_u[DD];
    if (tid < DD / 4)
        async_copy_b128(&s_u[tid * 4], &u[b * DD + tid * 4]);
    async_wait_all();
    __syncthreads();

    const int mrow = lane & 15;           // row of the 16-row A tile
    const int hi   = lane >> 4;           // 0 -> K={0,1}, 1 -> K={2,3}
    const int m0   = blockIdx.x * MTILE_LOGIT + wave * 16;
    const int m    = m0 + mrow;
    const int rt   = rel_time[b * MM + m];

    const float* kr = key_mems + ((size_t)(b * HH + hop) * MM + m) * DD;
    const float* tr = TA + ((size_t)hop * TT + rt) * DD;

    v8f c = {};
#pragma unroll 4
    for (int k0 = 0; k0 < DD; k0 += 4) {
        const int kk = k0 + (hi << 1);
        v2f a, bv;
        a[0]  = kr[kk]     + tr[kk];
        a[1]  = kr[kk + 1] + tr[kk + 1];
        bv[0] = s_u[kk];
        bv[1] = s_u[kk + 1];
        c = __builtin_amdgcn_wmma_f32_16x16x4_f32(
                /*neg_a=*/false, a, /*neg_b=*/false, bv,
                /*c_mod=*/(short)0, c, /*reuse_a=*/false, /*reuse_b=*/false);
    }

    // C layout: VGPR r holds M=r (lanes 0-15) and M=r+8 (lanes 16-31).
    // Column N=0 lives in lane 0 / lane 16 -> those lanes emit 8 logits each.
    if (mrow == 0) {
#pragma unroll
        for (int r = 0; r < 8; ++r)
            logits[(size_t)b * MM + m0 + hi * 8 + r] = c[r];
    }
}

// ---------------------------------------------------------------------------
// In-place softmax over M per batch; also zeroes the o accumulator.
// ---------------------------------------------------------------------------
__global__ __launch_bounds__(1024) void softmax_kernel(
        float* __restrict__ logits,
        float* __restrict__ o)
{
    const int b   = blockIdx.x;
    const int tid = threadIdx.x;
    __shared__ float red[1024];

    const float v0 = logits[(size_t)b * MM + tid];
    const float v1 = logits[(size_t)b * MM + tid + 1024];

    red[tid] = fmaxf(v0, v1);
    __syncthreads();
    for (int s = 512; s >= 1; s >>= 1) {
        if (tid < s) red[tid] = fmaxf(red[tid], red[tid + s]);
        __syncthreads();
    }
    const float gmax = red[0];
    __syncthreads();

    const float e0 = __expf(v0 - gmax);
    const float e1 = __expf(v1 - gmax);
    red[tid] = e0 + e1;
    __syncthreads();
    for (int s = 512; s >= 1; s >>= 1) {
        if (tid < s) red[tid] += red[tid + s];
        __syncthreads();
    }
    const float inv = 1.0f / red[0];

    logits[(size_t)b * MM + tid]        = e0 * inv;
    logits[(size_t)b * MM + tid + 1024] = e1 * inv;
    if (tid < DD) o[b * DD + tid] = 0.f;   // prep accumulator for out_kernel
}

// ---------------------------------------------------------------------------
// o[b,d] += sum_m p[b,m] * (val_mems[b,hop,m,d] + TC[hop, rel_time[b,m], d])
// 256 threads = 128 d-quads (float4) x 2 m-streams; partials merged with
// global_atomic_add_f32.
// ---------------------------------------------------------------------------
__global__ __launch_bounds__(256) void out_kernel(
        int hop,
        const int* __restrict__ rel_time,
        const float* __restrict__ val_mems,
        const float* __restrict__ TC,
        const float* __restrict__ p,
        float* __restrict__ o)
{
    const int b    = blockIdx.y;
    const int mbeg = blockIdx.x * MTILE_OUT;
    const int tid  = threadIdx.x;

    __shared__ float s_p[MTILE_OUT];
    __shared__ int   s_rt[MTILE_OUT];
    // Async-stage the probability tile and the time indices (1 KB each).
    if (tid < MTILE_OUT / 4) {
        async_copy_b128(&s_p[tid * 4], &p[(size_t)b * MM + mbeg + tid * 4]);
    } else if (tid < MTILE_OUT / 2) {
        const int t = tid - MTILE_OUT / 4;
        async_copy_b128(&s_rt[t * 4], &rel_time[b * MM + mbeg + t * 4]);
    }
    async_wait_all();
    __syncthreads();

    const int dq   = (tid & 127) * 4;     // d base of this thread's float4
    const int half = tid >> 7;            // 0/1 -> which 128 m's
    const int mloc = half * 128;

    float ax = 0.f, ay = 0.f, az = 0.f, aw = 0.f;
    const size_t valBase = (size_t)(b * HH + hop) * MM * DD;
#pragma unroll 4
    for (int mi = 0; mi < 128; ++mi) {
        const float pm  = s_p[mloc + mi];
        const float4 v  = *(const float4*)(val_mems + valBase +
                                           (size_t)(mbeg + mloc + mi) * DD + dq);
        const float4 t4 = *(const float4*)(TC + ((size_t)hop * TT + s_rt[mloc + mi]) * DD + dq);
        ax += pm * (v.x + t4.x);
        ay += pm * (v.y + t4.y);
        az += pm * (v.z + t4.z);
        aw += pm * (v.w + t4.w);
    }
    atomicAdd(&o[b * DD + dq + 0], ax);
    atomicAdd(&o[b * DD + dq + 1], ay);
    atomicAdd(&o[b * DD + dq + 2], az);
    atomicAdd(&o[b * DD + dq + 3], aw);
}

// ---------------------------------------------------------------------------
// u += o ; on the last hop also emit to d_out.
// ---------------------------------------------------------------------------
__global__ __launch_bounds__(DD) void update_kernel(
        const float* __restrict__ o,
        float* __restrict__ u,
        float* __restrict__ out,
        int last)
{
    const int b = blockIdx.x;
    const int d = threadIdx.x;
    const float nu = u[b * DD + d] + o[b * DD + d];
    u[b * DD + d] = nu;
    if (last) out[(size_t)b * DD + d] = nu;
}

// ---------------------------------------------------------------------------
extern "C" void kernel_launch(void* const* d_in, const int* in_sizes, int n_in,
                              void* d_out, int out_size, void* d_ws, size_t ws_size,
                              hipStream_t stream)
{
    const int*   query    = (const int*)d_in[0];
    const int*   rel_time = (const int*)d_in[1];
    const float* key_mems = (const float*)d_in[2];
    const float* val_mems = (const float*)d_in[3];
    const float* B_emb    = (const float*)d_in[4];
    const float* TA       = (const float*)d_in[5];
    const float* TC       = (const float*)d_in[6];
    const float* pos_enc  = (const float*)d_in[7];
    float* out = (float*)d_out;

    // Workspace layout (floats): u[B*D] | p[B*M] | o[B*D]  (~384 KB)
    float* u      = (float*)d_ws;
    float* logits = u + BB * DD;
    float* o      = logits + (size_t)BB * MM;

    embed_kernel<<<dim3(BB), dim3(DD), 0, stream>>>(query, B_emb, pos_enc, u);

    for (int hop = 0; hop < HH; ++hop) {
        logit_kernel<<<dim3(MM / MTILE_LOGIT, BB), dim3(256), 0, stream>>>(
            hop, rel_time, key_mems, TA, u, logits);
        softmax_kernel<<<dim3(BB), dim3(1024), 0, stream>>>(logits, o);
        out_kernel<<<dim3(MM / MTILE_OUT, BB), dim3(256), 0, stream>>>(
            hop, rel_time, val_mems, TC, logits, o);
        update_kernel<<<dim3(BB), dim3(DD), 0, stream>>>(o, u, out, hop == HH - 1);
    }
}